// SphericalHarmonics_12206297055386
// MI455X (gfx1250) — compile-verified
//
#include <hip/hip_runtime.h>
#include <cstdint>

// Spherical harmonics, L=20 -> 400 cols/point, B=250000 points.
// Memory-bound: 400MB of streamed output => ~17us floor @ 23.3 TB/s.
// Compute (~0.4 G fp32 ops) is ~3x under the floor, so no WMMA: the math is a
// per-point recurrence, and a matmul reformulation inflates FLOPs past the
// memory floor. CDNA5-specific path used instead: LDS-transposed output tiles
// drained with GLOBAL_STORE_ASYNC_FROM_LDS_B128 (ASYNCcnt), double-buffered so
// compute overlaps the async drain.

#define LMAX   20
#define NCOL   (LMAX * LMAX)      // 400
#define BLK    256                // threads per workgroup (8 waves @ wave32)
#define CHUNK  16                 // columns staged per flush; 400/16 = 25 exact
#define NCHUNK (NCOL / CHUNK)     // 25
#define STR    20                 // padded row stride (floats): 16B-aligned rows
                                  // for B128 LDS reads; 20*16%64==0 -> only a
                                  // 2-way bank conflict on scatter writes
#define BUFSZ  (BLK * STR)        // 5120 floats per buffer (40KB total LDS)

// K(l,m) = sqrt((2l+1)/(4pi)) * sqrt((l-m)!/(l+m)!)
//        = sqrt( (2l+1)/(4pi) / prod_{k=l-m+1}^{l+m} k )
// Called only with literal l,m under full unroll -> LLVM folds the loop and
// the sqrt-of-constant to an immediate.
__device__ __forceinline__ float kscale(int l, int m, bool sq2) {
    double prod = 1.0;
    for (int k = l - m + 1; k <= l + m; ++k) prod *= (double)k;
    double v = __builtin_sqrt((2.0 * l + 1.0) / (4.0 * 3.14159265358979323846) / prod);
    if (sq2) v *= 1.4142135623730951;
    return (float)v;
}

__global__ __launch_bounds__(BLK) void sh_kernel(const float* __restrict__ lonlat,
                                                 float* __restrict__ out, int B) {
    __shared__ float lbuf[2 * BUFSZ];            // double-buffered 256x16 tile

    const int tid    = (int)threadIdx.x;
    const int wgbase = (int)blockIdx.x * BLK;
    const int p      = wgbase + tid;
    const int pc     = (p < B) ? p : (B - 1);    // clamp tail loads

    const float D2R = 0.017453292519943295f;
    const float2 ll = ((const float2*)lonlat)[pc];
    const float phi = (ll.x + 180.0f) * D2R;
    const float the = (ll.y + 90.0f) * D2R;

    const float x  = __cosf(the);
    const float sx = __sinf(the);
    const float c1 = __cosf(phi);
    const float s1 = __sinf(phi);

    // cos(m*phi), sin(m*phi) via angle-addition recurrence (Chebyshev form)
    float cm[LMAX], sm[LMAX];
    cm[0] = 1.0f; sm[0] = 0.0f;
    cm[1] = c1;   sm[1] = s1;
    const float tc = 2.0f * c1;
#pragma unroll
    for (int m = 2; m < LMAX; ++m) {
        cm[m] = tc * cm[m - 1] - cm[m - 2];
        sm[m] = tc * sm[m - 1] - sm[m - 2];
    }

    const uint64_t outbase = (uint64_t)out;

    // Drain one 256-point x 16-col LDS tile with async LDS->global B128 stores.
    // Per lane: 4 float4s; per wave: 4 async instructions outstanding.
    // s_wait_asynccnt 4 => the *previous* flush (in-order completion) is done,
    // so after the barrier its buffer may be overwritten.
    auto flush = [&](int c) {
        __syncthreads();                          // tile writes visible to all waves
#pragma unroll
        for (int i = 0; i < 4; ++i) {
            const int idx = tid + BLK * i;        // float4 index within tile
            const int pl  = idx >> 2;             // point row 0..255
            const int c4  = idx & 3;              // float4 column 0..3
            const int pp  = wgbase + pl;
            if (pp < B) {
                // Derive the LDS byte offset from the *real* pointer: the
                // ptrtoint captures lbuf so the compiler cannot DSE the tile
                // writes feeding the asm; low 32 bits of a generic LDS pointer
                // are the hardware LDS byte address.
                unsigned ldsoff =
                    (unsigned)(uintptr_t)&lbuf[(c & 1) * BUFSZ + pl * STR + c4 * 4];
                int goff = (pp * NCOL + c * CHUNK + c4 * 4) * 4;
                asm volatile("global_store_async_from_lds_b128 %0, %1, %2"
                             :: "v"(goff), "v"(ldsoff), "s"(outbase)
                             : "memory");
            }
        }
        asm volatile("s_wait_asynccnt 4" ::: "memory");
        __syncthreads();                          // all waves' older stores retired
    };

    // col is always a compile-time constant at each call site (full unroll).
    auto emit = [&](int col, float v) {
        lbuf[((col >> 4) & 1) * BUFSZ + tid * STR + (col & (CHUNK - 1))] = v;
        if ((col & (CHUNK - 1)) == CHUNK - 1) flush(col >> 4);
    };

    // Associated Legendre recurrence, row-major in l so only 3 rows live.
    float Pl1[LMAX], Pl2[LMAX], Pc[LMAX];
#pragma unroll
    for (int m = 0; m < LMAX; ++m) { Pl1[m] = 0.0f; Pl2[m] = 0.0f; }
    Pl1[0] = 1.0f;                                // P[0,0]

    emit(0, kscale(0, 0, false) * Pl1[0]);        // l=0, m=0

#pragma unroll
    for (int l = 1; l < LMAX; ++l) {
        const float a = (float)(2 * l - 1);
        Pc[l]     = -a * sx * Pl1[l - 1];         // P[l,l]   = -(2l-1) sx P[l-1,l-1]
        Pc[l - 1] =  a * x  * Pl1[l - 1];         // P[l,l-1] =  (2l-1) x  P[l-1,l-1]
#pragma unroll
        for (int m = 0; m <= l - 2; ++m) {        // division folded into constants
            const float ca = (float)((2.0 * l - 1.0) / (double)(l - m));
            const float cb = (float)(-((double)(l + m - 1)) / (double)(l - m));
            Pc[m] = ca * x * Pl1[m] + cb * Pl2[m];
        }

        // Reference column order inside row l (base l*l):
        //   i=0..l-1 : m=-l+i -> sin(|m| phi), |m| = l-i (descending)
        //   i=l      : m=0
        //   j=1..l   : cos(j phi)
#pragma unroll
        for (int i = 0; i < l; ++i) {
            const int am = l - i;
            emit(l * l + i, kscale(l, am, true) * sm[am] * Pc[am]);
        }
        emit(l * l + l, kscale(l, 0, false) * Pc[0]);
#pragma unroll
        for (int j = 1; j <= l; ++j) {
            emit(l * l + l + j, kscale(l, j, true) * cm[j] * Pc[j]);
        }

#pragma unroll
        for (int m = 0; m <= l; ++m) { Pl2[m] = Pl1[m]; Pl1[m] = Pc[m]; }
    }
    // last emit was col 399 -> flush(24) already issued.
    // s_endpgm performs an implicit wait-idle, draining remaining async stores.
}

extern "C" void kernel_launch(void* const* d_in, const int* in_sizes, int n_in,
                              void* d_out, int out_size, void* d_ws, size_t ws_size,
                              hipStream_t stream) {
    (void)n_in; (void)out_size; (void)d_ws; (void)ws_size;
    const float* lonlat = (const float*)d_in[0];
    float*       out    = (float*)d_out;
    const int    B      = in_sizes[0] / 2;
    const int    grid   = (B + BLK - 1) / BLK;
    sh_kernel<<<grid, BLK, 0, stream>>>(lonlat, out, B);
}